// XLMRobertaForNERWithCRFLoRA_75084618269210
// MI455X (gfx1250) — compile-verified
//
#include <hip/hip_runtime.h>
#include <hip/hip_bf16.h>

// Problem constants
#define BB 64
#define SS 512
#define HH 768
#define LL 9
#define LP 16   // L padded to WMMA N=16

typedef float v2f __attribute__((ext_vector_type(2)));
typedef float v8f __attribute__((ext_vector_type(8)));

// ---------------------------------------------------------------------------
// Kernel A: emissions = hidden @ W_cls + b_cls   via V_WMMA_F32_16X16X4_F32
// Grid: 256 blocks x 256 threads (8 waves). Each wave owns a 16-row tile,
// block owns 128 rows. W_cls staged padded (768x16) in LDS (48 KB).
// ---------------------------------------------------------------------------
__global__ void emissions_wmma_kernel(const float* __restrict__ hidden,
                                      const float* __restrict__ Wc,
                                      const float* __restrict__ bc,
                                      float* __restrict__ emis) {
  __shared__ float ldsW[HH * LP];  // 48 KB
  const int tid = threadIdx.x;

  // Stage W_cls padded to 16 columns (cols 9..15 = 0)
  for (int idx = tid; idx < HH * LP; idx += 256) {
    const int k = idx >> 4;
    const int n = idx & 15;
    ldsW[idx] = (n < LL) ? Wc[k * LL + n] : 0.0f;
  }
  __syncthreads();

  const int wave = tid >> 5;       // 0..7
  const int lane = tid & 31;
  const int half = lane >> 4;      // 0: K=0,1  1: K=2,3
  const int l16  = lane & 15;      // row within tile (A) / column N (B,D)

  const int tileBase = blockIdx.x * 128 + wave * 16;
  const float* __restrict__ hrow = hidden + (size_t)(tileBase + l16) * HH;

  v8f acc = {0.f, 0.f, 0.f, 0.f, 0.f, 0.f, 0.f, 0.f};

#pragma unroll 4
  for (int k = 0; k < HH; k += 4) {
    const int kk = k + half * 2;
    // A fragment: lane<16 -> (K=k+0, k+1) of row l16 ; lane>=16 -> (k+2, k+3)
    const float2 av = *reinterpret_cast<const float2*>(hrow + kk);
    v2f a; a[0] = av.x; a[1] = av.y;
    // B fragment: v0: row K=kk, col l16 ; v1: row K=kk+1, col l16
    v2f b; b[0] = ldsW[kk * LP + l16]; b[1] = ldsW[(kk + 1) * LP + l16];
    acc = __builtin_amdgcn_wmma_f32_16x16x4_f32(
        /*neg_a=*/false, a, /*neg_b=*/false, b,
        /*c_mod=*/(short)0, acc, /*reuse_a=*/false, /*reuse_b=*/false);
  }

  // D layout: VGPR r -> (M=r, N=lane) for lanes 0-15, (M=r+8, N=lane-16) hi
  if (l16 < LL) {
    const float bias = bc[l16];
#pragma unroll
    for (int r = 0; r < 8; ++r) {
      const int M = r + half * 8;
      emis[(size_t)(tileBase + M) * LL + l16] = acc[r] + bias;
    }
  }
}

// ---------------------------------------------------------------------------
// Kernel B: CRF forward (logZ), numerator, viterbi + in-LDS backtrack.
// Grid: 64 blocks (one per batch) x 32 threads (one wave32). Lane j < 9 owns
// tag state j. Scores broadcast with compile-time-lane __shfl (v_readlane).
// ---------------------------------------------------------------------------
__global__ void crf_kernel(const float* __restrict__ emis,
                           const float* __restrict__ startT,
                           const float* __restrict__ endT,
                           const float* __restrict__ trans,
                           const int* __restrict__ att,
                           const int* __restrict__ labels,
                           float* __restrict__ llh_out,   // [B]
                           float* __restrict__ preds) {   // [B*S] as float
  const int b = blockIdx.x;
  const int lane = threadIdx.x;

  __shared__ float sT[LL * LL];
  __shared__ unsigned char hist[SS * LL];  // backpointers, t = 1..511

  for (int i = lane; i < LL * LL; i += 32) sT[i] = trans[i];
  __syncthreads();

  const float* __restrict__ eb = emis + (size_t)b * SS * LL;
  const int* __restrict__ ab = att + b * SS;
  const int* __restrict__ lb = labels + b * SS;

  const int j = (lane < LL) ? lane : 0;  // clamp idle lanes
  float tcol[LL];
#pragma unroll
  for (int i = 0; i < LL; ++i) tcol[i] = sT[i * LL + j];  // column j of T
  const float endj = endT[j];

  // ---- length + numerator: parallel over t, butterfly reduce ----
  int lenPart = 0;
  float numPart = 0.0f;
  for (int t = lane; t < SS; t += 32) {
    const int m = (t == 0) ? 1 : (ab[t] != 0);  // mask[:,0] forced true
    lenPart += m;
    if (t == 0) {
      const int tg0 = lb[0];
      numPart += startT[tg0] + eb[tg0];
    } else if (m) {
      const int tgp = lb[t - 1];  // mask contiguous -> prev is valid
      const int tgc = lb[t];
      numPart += sT[tgp * LL + tgc] + eb[t * LL + tgc];
    }
  }
#pragma unroll
  for (int off = 16; off >= 1; off >>= 1) {
    numPart += __shfl_xor(numPart, off, 32);
    lenPart += __shfl_xor(lenPart, off, 32);
  }
  const int seqEnd = lenPart - 1;
  const float numerator = numPart + endT[lb[seqEnd]];

  // ---- forward scan (logsumexp) + viterbi (max/argmax) ----
  float score  = startT[j] + eb[j];  // t = 0
  float scoreV = score;

  for (int t = 1; t < SS; ++t) {
    const float ej = eb[t * LL + j];
    const int m = (ab[t] != 0);

    float sArr[LL], svArr[LL];
#pragma unroll
    for (int i = 0; i < LL; ++i) {
      sArr[i]  = __shfl(score,  i, 32);
      svArr[i] = __shfl(scoreV, i, 32);
    }
    // logsumexp over i of (score_i + T[i][j]) + emis
    float mx = sArr[0] + tcol[0];
#pragma unroll
    for (int i = 1; i < LL; ++i) mx = fmaxf(mx, sArr[i] + tcol[i]);
    float sum = 0.0f;
#pragma unroll
    for (int i = 0; i < LL; ++i) sum += __expf(sArr[i] + tcol[i] - mx);
    const float nxt = mx + __logf(sum) + ej;

    // viterbi: max / first-argmax over i
    float bv = svArr[0] + tcol[0];
    int bi = 0;
#pragma unroll
    for (int i = 1; i < LL; ++i) {
      const float c = svArr[i] + tcol[i];
      if (c > bv) { bv = c; bi = i; }
    }
    const float nxtV = bv + ej;

    if (lane < LL) hist[t * LL + j] = (unsigned char)bi;
    score  = m ? nxt  : score;
    scoreV = m ? nxtV : scoreV;
  }

  // ---- logZ and last_tag (all lanes replicate the tiny reductions) ----
  const float finS = score + endj;
  const float finV = scoreV + endj;
  float fsArr[LL], fvArr[LL];
#pragma unroll
  for (int i = 0; i < LL; ++i) {
    fsArr[i] = __shfl(finS, i, 32);
    fvArr[i] = __shfl(finV, i, 32);
  }
  float mx = fsArr[0];
#pragma unroll
  for (int i = 1; i < LL; ++i) mx = fmaxf(mx, fsArr[i]);
  float sum = 0.0f;
#pragma unroll
  for (int i = 0; i < LL; ++i) sum += __expf(fsArr[i] - mx);
  const float logZ = mx + __logf(sum);

  float bestV = fvArr[0];
  int lastTag = 0;
#pragma unroll
  for (int i = 1; i < LL; ++i) {
    if (fvArr[i] > bestV) { bestV = fvArr[i]; lastTag = i; }
  }

  __syncthreads();  // hist visible (single-wave block: barrier ~ s_nop + waits)

  if (lane == 0) {
    llh_out[b] = numerator - logZ;

    // backtrack from LDS
    float* __restrict__ pb = preds + (size_t)b * SS;
    int tag = lastTag;
    pb[SS - 1] = (ab[SS - 1] != 0) ? (float)tag : 0.0f;
    for (int t = SS - 1; t >= 1; --t) {
      const int m = (ab[t] != 0);
      const int prev = m ? (int)hist[t * LL + tag] : tag;
      const int mprev = (t - 1 == 0) ? 1 : (ab[t - 1] != 0);
      pb[t - 1] = mprev ? (float)prev : 0.0f;
      tag = prev;
    }
  }
}

// ---------------------------------------------------------------------------
// Kernel C: deterministic fixed-order loss reduction over 64 llh values.
// ---------------------------------------------------------------------------
__global__ void loss_reduce_kernel(const float* __restrict__ llh,
                                   float* __restrict__ loss) {
  if (threadIdx.x == 0 && blockIdx.x == 0) {
    float s = 0.0f;
    for (int b = 0; b < BB; ++b) s += llh[b];
    *loss = -s * (1.0f / (float)BB);
  }
}

// ---------------------------------------------------------------------------
extern "C" void kernel_launch(void* const* d_in, const int* in_sizes, int n_in,
                              void* d_out, int out_size, void* d_ws, size_t ws_size,
                              hipStream_t stream) {
  const float* hidden = (const float*)d_in[0];
  const float* Wc     = (const float*)d_in[1];
  const float* bc     = (const float*)d_in[2];
  const float* startT = (const float*)d_in[3];
  const float* endT   = (const float*)d_in[4];
  const float* trans  = (const float*)d_in[5];
  const int*   att    = (const int*)d_in[6];
  const int*   labels = (const int*)d_in[7];

  float* out   = (float*)d_out;
  float* emis  = out;                      // B*S*L = 294912
  float* loss  = out + (BB * SS * LL);     // 1
  float* preds = loss + 1;                 // B*S = 32768
  float* llh_ws = (float*)d_ws;            // 64 floats of scratch

  // GEMM: 32768 rows / 128 rows per block = 256 blocks of 256 threads
  emissions_wmma_kernel<<<256, 256, 0, stream>>>(hidden, Wc, bc, emis);
  // CRF: one wave32 per batch element
  crf_kernel<<<BB, 32, 0, stream>>>(emis, startT, endT, trans, att, labels,
                                    llh_ws, preds);
  loss_reduce_kernel<<<1, 32, 0, stream>>>(llh_ws, loss);
}